// trunc_clip_3762391352096
// MI455X (gfx1250) — compile-verified
//
#include <hip/hip_runtime.h>
#include <stdint.h>

// ---------------------------------------------------------------------------
// Zero the top-32 and bottom-32 elements of each 8192-float row, copy the rest.
// Memory-bound: 512 MB total traffic -> ~22us floor @ 23.3 TB/s on MI455X.
// Strategy: TDM (tensor_load_to_lds) DMA of one row per workgroup into LDS,
// register-resident 4-pass radix select (both tails at once) with wave32
// ballot-aggregated LDS histogram atomics, single coalesced b128 store pass.
// ---------------------------------------------------------------------------

#define ROWLEN 8192
#define BLOCK  512
#define PER_T  (ROWLEN / BLOCK)   // 16 elements per thread
#define KSEL   32

#if __has_builtin(__builtin_amdgcn_tensor_load_to_lds)
#define HAVE_TDM 1
#else
#define HAVE_TDM 0
#endif

typedef unsigned int u32x4_t __attribute__((ext_vector_type(4)));
typedef int          i32x8_t __attribute__((ext_vector_type(8)));
typedef int          i32x4_t __attribute__((ext_vector_type(4)));

// Order-preserving float-bits <-> uint map (larger float <=> larger uint).
__device__ __forceinline__ unsigned f2ord(unsigned b) {
  return b ^ (unsigned)(((int)b >> 31) | (int)0x80000000);
}
__device__ __forceinline__ unsigned ord2f(unsigned u) {
  return (u & 0x80000000u) ? (u ^ 0x80000000u) : ~u;
}

// Wave32 ballot-aggregated histogram increment: one ds_add per distinct bin
// per wave instead of one per lane (defeats exponent-clustering conflicts).
__device__ __forceinline__ void hist_ballot_add(unsigned* h, unsigned bin, bool active) {
  unsigned act = (unsigned)__ballot(active);        // wave32: low 32 bits
  while (act) {
    const int leader = __ffs(act) - 1;
    const unsigned lbin = (unsigned)__shfl((int)bin, leader);
    const unsigned same = (unsigned)__ballot(active && (bin == lbin));
    if ((int)(threadIdx.x & 31u) == leader)
      atomicAdd(&h[lbin], (unsigned)__popc(same));
    act &= ~same;
  }
}

__global__ __launch_bounds__(BLOCK) void rowmask_topk_kernel(
    const float* __restrict__ x, float* __restrict__ out) {
  __shared__ float    rowbuf[ROWLEN];   // 32 KB row staging (TDM landing zone)
  __shared__ unsigned hist[512];        // [0..255] top-side, [256..511] bottom-side
  __shared__ unsigned ctrl[8];          // 0:pre_hi 1:k_hi 2:pre_lo 3:k_lo 4/5:eq ctrs

  const unsigned tid = threadIdx.x;
  const size_t rowbase = (size_t)blockIdx.x * ROWLEN;
  const float* src = x + rowbase;
  float* dst = out + rowbase;

  if (tid == 0) {
    ctrl[0] = 0u; ctrl[1] = KSEL;
    ctrl[2] = 0u; ctrl[3] = KSEL;
    ctrl[4] = 0u; ctrl[5] = 0u;
  }

#if HAVE_TDM
  // --- Tensor Data Mover: DMA the whole 32KB row global->LDS (wave 0 issues).
  if (tid < 32) {
    const uint64_t ga = (uint64_t)(uintptr_t)src;
    const unsigned lds_base = (unsigned)(uintptr_t)(void*)&rowbuf[0];
    // D# group0: count=1 | lds_addr | global_addr[56:0] | type=2 (image)
    u32x4_t g0 = {
      0x1u,
      lds_base,
      (unsigned)(ga & 0xFFFFFFFFull),
      (unsigned)((ga >> 32) & 0x01FFFFFFull) | (2u << 30)
    };
    // D# group1: data_size=4B; tensor_dim0=8192, tensor_dim1=1;
    //            tile_dim0=8192, tile_dim1=1; dim0_stride=8192.
    i32x8_t g1 = {
      (int)(2u << 16),                              // data_size=2 (4 bytes)
      (int)((unsigned)(ROWLEN & 0xFFFF) << 16),     // tensor_dim0[15:0]
      (int)(1u << 16),                              // tensor_dim1 = 1
      (int)((unsigned)(ROWLEN & 0xFFFF) << 16),     // tile_dim0 = 8192
      1,                                            // tile_dim1 = 1, tile_dim2 = 0
      (int)ROWLEN,                                  // tensor_dim0_stride lo32
      (int)((unsigned)(ROWLEN & 0xFFFF) << 16),     // tensor_dim1_stride[15:0]
      0
    };
    // Higher dims are degenerate (length 1 / tiles unused).
    i32x4_t g2 = { 1, 1, (int)ROWLEN, 0 };          // dim2=1, dim3=1, dim2_stride
    i32x4_t g3 = { (int)ROWLEN, (int)(1u << 16), 0, 0 }; // dim3_stride, dim4=1
    i32x8_t g4 = { 0, 0, 0, 0, 0, 0, 0, 0 };        // trailing operand (unused)
    __builtin_amdgcn_tensor_load_to_lds(g0, g1, g2, g3, g4, 0);
#if __has_builtin(__builtin_amdgcn_s_wait_tensorcnt)
    __builtin_amdgcn_s_wait_tensorcnt((short)0);
#else
    asm volatile("s_wait_tensorcnt 0" ::: "memory");
#endif
  }
#else
  // Fallback: cooperative coalesced b128 copy global->LDS.
  for (int i = 0; i < PER_T / 4; ++i) {
    const unsigned idx4 = tid + (unsigned)i * BLOCK;
    ((float4*)rowbuf)[idx4] = ((const float4*)src)[idx4];
  }
#endif
  __syncthreads();

  // Pull this thread's 16 elements out of LDS once, keep as ordered uints in VGPRs.
  unsigned u[PER_T];
  for (int i = 0; i < PER_T / 4; ++i) {
    const unsigned idx4 = tid + (unsigned)i * BLOCK;
    const float4 v = ((const float4*)rowbuf)[idx4];
    u[4*i+0] = f2ord(__float_as_uint(v.x));
    u[4*i+1] = f2ord(__float_as_uint(v.y));
    u[4*i+2] = f2ord(__float_as_uint(v.z));
    u[4*i+3] = f2ord(__float_as_uint(v.w));
  }

  // 4-pass 256-way radix select: find the 32nd-largest (thr_hi) and
  // 32nd-smallest (thr_lo) ordered-uint values simultaneously.
  for (int p = 0; p < 4; ++p) {
    const int shift = 24 - 8 * p;
    const unsigned msk = p ? (0xFFFFFFFFu << (shift + 8)) : 0u;
    const unsigned pre_hi = ctrl[0], k_hi = ctrl[1];
    const unsigned pre_lo = ctrl[2], k_lo = ctrl[3];
    hist[tid] = 0u;
    __syncthreads();

    for (int e = 0; e < PER_T; ++e) {
      const unsigned ue = u[e];
      const unsigned b = (ue >> shift) & 0xFFu;
      hist_ballot_add(hist,       b, (ue & msk) == pre_hi);
      hist_ballot_add(hist + 256, b, (ue & msk) == pre_lo);
    }
    __syncthreads();

    // Hillis-Steele: suffix-sum bins 0..255 (top side), prefix-sum 256..511 (bottom).
    for (int off = 1; off < 256; off <<= 1) {
      unsigned add = 0u;
      if (tid < 256u) {
        if (tid + off < 256u) add = hist[tid + off];
      } else {
        const unsigned b = tid - 256u;
        if (b >= (unsigned)off) add = hist[256u + b - off];
      }
      __syncthreads();
      hist[tid] += add;
      __syncthreads();
    }

    // Exactly one thread per side satisfies the bracketing condition.
    if (tid < 256u) {
      const unsigned S  = hist[tid];
      const unsigned S1 = (tid < 255u) ? hist[tid + 1] : 0u;
      if (S >= k_hi && S1 < k_hi) {
        ctrl[0] = pre_hi | (tid << shift);
        ctrl[1] = k_hi - S1;                // rank within selected bin
      }
    } else {
      const unsigned b  = tid - 256u;
      const unsigned P  = hist[tid];
      const unsigned P0 = b ? hist[tid - 1] : 0u;
      if (P >= k_lo && P0 < k_lo) {
        ctrl[2] = pre_lo | (b << shift);
        ctrl[3] = k_lo - P0;
      }
    }
    __syncthreads();
  }

  const unsigned thr_hi = ctrl[0], bud_hi = ctrl[1];  // bud = #equal elems to zero
  const unsigned thr_lo = ctrl[2], bud_lo = ctrl[3];

  // Decide + store. Strictly beyond a threshold always zeroes (K - bud of them);
  // elements equal to a threshold consume a budget so exactly K zero per side.
  for (int i = 0; i < PER_T / 4; ++i) {
    const unsigned idx4 = tid + (unsigned)i * BLOCK;
    float4 o;
    float* po = reinterpret_cast<float*>(&o);
    #pragma unroll
    for (int c = 0; c < 4; ++c) {
      const unsigned ue = u[4*i + c];
      bool zero = (ue > thr_hi) || (ue < thr_lo);
      if (ue == thr_hi) zero |= (atomicAdd(&ctrl[4], 1u) < bud_hi);
      if (ue == thr_lo) zero |= (atomicAdd(&ctrl[5], 1u) < bud_lo);
      po[c] = zero ? 0.0f : __uint_as_float(ord2f(ue));   // bit-exact passthrough
    }
    ((float4*)dst)[idx4] = o;
  }
}

extern "C" void kernel_launch(void* const* d_in, const int* in_sizes, int n_in,
                              void* d_out, int out_size, void* d_ws, size_t ws_size,
                              hipStream_t stream) {
  (void)n_in; (void)out_size; (void)d_ws; (void)ws_size;
  const float* x = (const float*)d_in[0];
  float* out = (float*)d_out;
  const int rows = in_sizes[0] / ROWLEN;   // 8192
  rowmask_topk_kernel<<<dim3(rows), dim3(BLOCK), 0, stream>>>(x, out);
}